// DynamicFrustumMemory_4767413698787
// MI455X (gfx1250) — compile-verified
//
#include <hip/hip_runtime.h>

#define BATCH 8
#define SEQ   4096
#define ATOK  256
#define DIM   1024
#define LCOMB (ATOK + SEQ)   // 4352
#define KEEPN 256
#define NDROP (LCOMB - KEEPN) // 4096
#define MPAD  16             // ctx padded to one full WMMA M-tile

typedef __attribute__((ext_vector_type(2))) float v2f;
typedef __attribute__((ext_vector_type(8))) float v8f;

// ---------------------------------------------------------------------------
// 1) salience[b,l] = dot(combined[b,l,:], w_sal)   (combined = [active | x])
// ---------------------------------------------------------------------------
__global__ __launch_bounds__(256) void salience_kernel(
    const float* __restrict__ x, const float* __restrict__ active,
    const float* __restrict__ w_sal, float* __restrict__ sal) {
  const int token = blockIdx.x;            // 0 .. BATCH*LCOMB-1
  const int b = token / LCOMB;
  const int l = token % LCOMB;
  const float* src = (l < ATOK)
      ? (active + ((size_t)b * ATOK + l) * DIM)
      : (x + ((size_t)b * SEQ + (l - ATOK)) * DIM);
  __shared__ float red[256];
  float s = 0.f;
  for (int d = threadIdx.x; d < DIM; d += 256) s += src[d] * w_sal[d];
  red[threadIdx.x] = s;
  __syncthreads();
  for (int off = 128; off > 0; off >>= 1) {
    if ((int)threadIdx.x < off) red[threadIdx.x] += red[threadIdx.x + off];
    __syncthreads();
  }
  if (threadIdx.x == 0) sal[token] = red[0];
}

// ---------------------------------------------------------------------------
// 2) total[b,d] = sum_l combined[b,l,d]   (coalesced across d)
// ---------------------------------------------------------------------------
__global__ __launch_bounds__(256) void colsum_kernel(
    const float* __restrict__ x, const float* __restrict__ active,
    float* __restrict__ total) {
  const int b = blockIdx.y;
  const int d = blockIdx.x * 256 + threadIdx.x;
  float s = 0.f;
  const float* ab = active + (size_t)b * ATOK * DIM + d;
  for (int l = 0; l < ATOK; ++l) s += ab[(size_t)l * DIM];
  const float* xb = x + (size_t)b * SEQ * DIM + d;
  for (int l = 0; l < SEQ; ++l) s += xb[(size_t)l * DIM];
  total[b * DIM + d] = s;
}

// ---------------------------------------------------------------------------
// 3) top-k per batch via rank-by-count; output indices sorted ascending.
//    Tie-break matches lax.top_k: higher value first, lower index on ties.
// ---------------------------------------------------------------------------
__global__ __launch_bounds__(1024) void topk_kernel(
    const float* __restrict__ sal, int* __restrict__ top_idx) {
  const int b = blockIdx.x;
  __shared__ float s[LCOMB];
  __shared__ unsigned char keep[LCOMB];
  for (int i = threadIdx.x; i < LCOMB; i += 1024) s[i] = sal[b * LCOMB + i];
  __syncthreads();
  for (int i = threadIdx.x; i < LCOMB; i += 1024) {
    const float si = s[i];
    int rank = 0;
    for (int j = 0; j < LCOMB; ++j) {
      const float sj = s[j];
      rank += ((sj > si) || (sj == si && j < i)) ? 1 : 0;
    }
    keep[i] = (rank < KEEPN) ? 1 : 0;
  }
  __syncthreads();
  for (int i = threadIdx.x; i < LCOMB; i += 1024) {
    if (keep[i]) {
      int pos = 0;
      for (int j = 0; j < i; ++j) pos += keep[j];
      top_idx[b * KEEPN + pos] = i;
    }
  }
}

// ---------------------------------------------------------------------------
// 4) gather kept tokens -> new_active output section
// ---------------------------------------------------------------------------
__global__ __launch_bounds__(256) void gather_kernel(
    const float* __restrict__ x, const float* __restrict__ active,
    const int* __restrict__ top_idx, float* __restrict__ new_active) {
  const size_t g = (size_t)blockIdx.x * 256 + threadIdx.x; // BATCH*KEEPN*DIM
  const int d = (int)(g % DIM);
  const int k = (int)((g / DIM) % KEEPN);
  const int b = (int)(g / ((size_t)DIM * KEEPN));
  const int l = top_idx[b * KEEPN + k];
  const float v = (l < ATOK)
      ? active[((size_t)b * ATOK + l) * DIM + d]
      : x[((size_t)b * SEQ + (l - ATOK)) * DIM + d];
  new_active[g] = v;
}

// ---------------------------------------------------------------------------
// 5) kept_sum[b,d] = sum_k new_active[b,k,d]
// ---------------------------------------------------------------------------
__global__ __launch_bounds__(256) void keptsum_kernel(
    const float* __restrict__ new_active, float* __restrict__ kept_sum) {
  const int b = blockIdx.y;
  const int d = blockIdx.x * 256 + threadIdx.x;
  float s = 0.f;
  const float* nb = new_active + (size_t)b * KEEPN * DIM + d;
  for (int k = 0; k < KEEPN; ++k) s += nb[(size_t)k * DIM];
  kept_sum[b * DIM + d] = s;
}

// ---------------------------------------------------------------------------
// 6) cold_new / ctx.  ctx is written PADDED to 16 rows (rows 8..15 = 0) so
//    the WMMA kernel never needs a lane-divergent guard on the A operand.
// ---------------------------------------------------------------------------
__global__ __launch_bounds__(256) void ctx_kernel(
    const float* __restrict__ total, const float* __restrict__ kept_sum,
    const float* __restrict__ cold, float* __restrict__ cold_out,
    float* __restrict__ ctx_pad) {
  const int g = blockIdx.x * 256 + threadIdx.x;   // 0 .. MPAD*DIM-1
  if (g < BATCH * DIM) {
    const float ks = kept_sum[g];
    const float dropped_mean = (total[g] - ks) * (1.0f / (float)NDROP);
    const float cn = 0.9f * cold[g] + 0.1f * dropped_mean;
    cold_out[g] = cn;
    ctx_pad[g] = ks * (1.0f / (float)KEEPN) + cn;
  } else {
    ctx_pad[g] = 0.0f;                            // zero-pad rows 8..15
  }
}

// ---------------------------------------------------------------------------
// 7) proj[b,e] = sum_d ctx[b,d] * w_read[e,d]  via V_WMMA_F32_16X16X4_F32.
//    One wave per 16-wide N tile; M = 16 padded batch rows (8 real).
//    A (16x4 f32): lanes 0-15 hold K = {k0,k0+1}, lanes 16-31 K = {k0+2,k0+3}
//    B (4x16 f32): VGPR0 = rows K {k0 | k0+2}, VGPR1 = rows {k0+1 | k0+3}.
//    Branch-free inner loop: 2x global_load_b64 + 1x wmma per K-step.
// ---------------------------------------------------------------------------
__global__ __launch_bounds__(32) void proj_wmma_kernel(
    const float* __restrict__ ctx_pad, const float* __restrict__ w_read,
    float* __restrict__ proj) {
  const int lane  = threadIdx.x;       // 0..31
  const int n0    = blockIdx.x * 16;   // output column tile
  const int half  = lane >> 4;         // 0 or 1
  const int row   = lane & 15;         // M index (A) / N index (B)
  const int khalf = half * 2;          // K sub-offset: 0 or 2
  const int n     = n0 + row;

  const float* aptr = ctx_pad + (size_t)row * DIM + khalf;   // stride 4 floats
  const float* bptr = w_read + (size_t)n * DIM + khalf;      // stride 4 floats

  v8f c = {};
#pragma unroll 8
  for (int k0 = 0; k0 < DIM; k0 += 4) {
    const v2f a  = *(const v2f*)(aptr + k0);   // one global_load_b64
    const v2f bm = *(const v2f*)(bptr + k0);   // one global_load_b64
    c = __builtin_amdgcn_wmma_f32_16x16x4_f32(
        /*neg_a=*/false, a, /*neg_b=*/false, bm,
        /*c_mod=*/(short)0, c, /*reuse_a=*/false, /*reuse_b=*/false);
  }
  // C/D layout: VGPR r, lanes 0-15 => M=r, N=lane; lanes 16-31 => M=r+8 (pad)
  if (lane < 16) {
#pragma unroll
    for (int r = 0; r < 8; ++r) {
      proj[r * DIM + n0 + lane] = c[r];
    }
  }
}

// ---------------------------------------------------------------------------
// 8) x_out = x + proj[b,:] broadcast over L (float4 vectorized)
// ---------------------------------------------------------------------------
__global__ __launch_bounds__(256) void add_kernel(
    const float* __restrict__ x, const float* __restrict__ proj,
    float* __restrict__ x_out) {
  const size_t g4 = (size_t)blockIdx.x * 256 + threadIdx.x; // elems/4
  const size_t g  = g4 * 4;
  const int b = (int)(g / ((size_t)SEQ * DIM));
  const int d = (int)(g % DIM);
  const float4 xv = ((const float4*)x)[g4];
  const float4 pv = *(const float4*)(proj + b * DIM + d);
  float4 o;
  o.x = xv.x + pv.x; o.y = xv.y + pv.y;
  o.z = xv.z + pv.z; o.w = xv.w + pv.w;
  ((float4*)x_out)[g4] = o;
}

// ---------------------------------------------------------------------------
extern "C" void kernel_launch(void* const* d_in, const int* in_sizes, int n_in,
                              void* d_out, int out_size, void* d_ws, size_t ws_size,
                              hipStream_t stream) {
  const float* x      = (const float*)d_in[0];  // [8,4096,1024]
  const float* active = (const float*)d_in[1];  // [8,256,1024]
  const float* cold   = (const float*)d_in[2];  // [8,1024]
  const float* w_sal  = (const float*)d_in[3];  // [1024]
  const float* w_read = (const float*)d_in[4];  // [1024,1024]

  float* out        = (float*)d_out;
  float* x_out      = out;                                        // [8,4096,1024]
  float* new_active = out + (size_t)BATCH * SEQ * DIM;            // [8,256,1024]
  float* cold_out   = new_active + (size_t)BATCH * KEEPN * DIM;   // [8,1024]

  float* ws       = (float*)d_ws;
  float* sal      = ws;                        // BATCH*LCOMB
  float* total    = sal + BATCH * LCOMB;       // BATCH*DIM
  float* kept_sum = total + BATCH * DIM;       // BATCH*DIM
  float* ctx_pad  = kept_sum + BATCH * DIM;    // MPAD*DIM (rows 8..15 zero)
  float* proj     = ctx_pad + MPAD * DIM;      // BATCH*DIM
  int*   idx      = (int*)(proj + BATCH * DIM);// BATCH*KEEPN

  salience_kernel<<<BATCH * LCOMB, 256, 0, stream>>>(x, active, w_sal, sal);
  colsum_kernel<<<dim3(DIM / 256, BATCH), 256, 0, stream>>>(x, active, total);
  topk_kernel<<<BATCH, 1024, 0, stream>>>(sal, idx);
  gather_kernel<<<(BATCH * KEEPN * DIM) / 256, 256, 0, stream>>>(x, active, idx, new_active);
  keptsum_kernel<<<dim3(DIM / 256, BATCH), 256, 0, stream>>>(new_active, kept_sum);
  ctx_kernel<<<(MPAD * DIM) / 256, 256, 0, stream>>>(total, kept_sum, cold, cold_out, ctx_pad);
  proj_wmma_kernel<<<DIM / 16, 32, 0, stream>>>(ctx_pad, w_read, proj);
  add_kernel<<<((size_t)BATCH * SEQ * DIM / 4) / 256, 256, 0, stream>>>(x, proj, x_out);
}